// HamiltonianNeuralODE_4750233829420
// MI455X (gfx1250) — compile-verified
//
#include <hip/hip_runtime.h>

// MI455X (gfx1250) Hamiltonian neural-ODE integrator, v2.
// Compute-bound (~172 GFLOP vs 128MB I/O) -> V_WMMA_F32_16X16X4_F32 path.
// v2 changes: p state lives in per-wave LDS (frees 64 VGPRs -> no spills),
// HIDDEN chunked 8x64 so total LDS = 147KB (<160KB -> 2 workgroups/WGP),
// W1 staging buffers first in LDS so DS loads use 16-bit immediates.

typedef __attribute__((ext_vector_type(2))) float v2f;
typedef __attribute__((ext_vector_type(8))) float v8f;

#define B_TOTAL 65536
#define ND 128
#define NH 512
#define CHUNK 64
#define NCHUNK (NH / CHUNK)          // 8
#define NTILES (CHUNK / 16)          // 4
#define NSTEPS 10
#define DT 0.1f
#define ROWS_PER_BLOCK 128           // 8 waves * 16 rows
#define PSTRIDE 132                  // padded row stride for pbuf (bank-safe)

struct SmemT {
  float w1f[ND * CHUNK];             // 8192: fwd layout (k>>2)*256 + n*4 + (k&3)
  float w1t[CHUNK * ND];             // 8192: bwd layout (kh>>2)*512 + nd*4 + (kh&3)
  float pbuf[8][16 * PSTRIDE];       // per-wave momentum state p (16 rows x 128)
  float b1[NH];
  float w2[NH];
  float scr[8][16 * 20];             // per-wave 16x16 transpose scratch
};

__global__ __launch_bounds__(256) void hnode_kernel(
    const float* __restrict__ x, const float* __restrict__ W1,
    const float* __restrict__ b1g, const float* __restrict__ W2g,
    float* __restrict__ out) {
  __shared__ SmemT s;
  const int tid  = threadIdx.x;
  const int wave = tid >> 5;
  const int lane = tid & 31;
  const int hi16 = lane >> 4;        // lane-half selects K/M sub-half (ISA layout)
  const int l16  = lane & 15;
  const int row  = blockIdx.x * ROWS_PER_BLOCK + wave * 16 + l16;

  for (int i = tid; i < NH; i += 256) { s.b1[i] = b1g[i]; s.w2[i] = W2g[i]; }

  float* pb  = s.pbuf[wave];
  float* scr = s.scr[wave];

  // q in registers (WMMA A-layout: frag kk = dims kk*4 + hi16*2 + {0,1}),
  // p in per-wave LDS (same logical layout, addressed per fragment).
  v2f qA[32];
  const float* xr = x + (size_t)row * (2 * ND);
  #pragma unroll
  for (int kk = 0; kk < 32; ++kk) {
    qA[kk] = *(const v2f*)(xr + kk * 4 + hi16 * 2);
    v2f pv = *(const v2f*)(xr + ND + kk * 4 + hi16 * 2);
    *(v2f*)&pb[l16 * PSTRIDE + kk * 4 + hi16 * 2] = pv;
  }

  for (int step = 0; step < NSTEPS; ++step) {
    v8f dacc[8];  // dV/dq accumulators, C-layout, 8 tiles of 16 dims
    #pragma unroll
    for (int d = 0; d < 8; ++d) dacc[d] = (v8f){0.f,0.f,0.f,0.f,0.f,0.f,0.f,0.f};

    for (int c = 0; c < NCHUNK; ++c) {
      const int hbase = c * CHUNK;
      __syncthreads();  // previous chunk's consumers done
      // Stage W1 chunk in both operand layouts (coalesced global reads).
      for (int e = tid; e < ND * CHUNK; e += 256) {
        int k = e >> 6, n = e & 63;
        s.w1f[(k >> 2) * 256 + n * 4 + (k & 3)] = W1[k * NH + hbase + n];
      }
      for (int e = tid; e < ND * CHUNK; e += 256) {
        int nd = e >> 6, kh = e & 63;
        s.w1t[(kh >> 2) * 512 + nd * 4 + (kh & 3)] = W1[nd * NH + hbase + kh];
      }
      __syncthreads();

      for (int nt = 0; nt < NTILES; ++nt) {    // hidden-col tiles of 16
        const int hn = hbase + nt * 16;
        const float bv = s.b1[hn + l16];
        v8f acc = {bv, bv, bv, bv, bv, bv, bv, bv};
        // Forward tile: h_acc = q @ W1[:, hn:hn+16] + b1
        #pragma unroll
        for (int kk = 0; kk < 32; ++kk) {
          v2f b = *(const v2f*)&s.w1f[kk * 256 + (nt * 16 + l16) * 4 + hi16 * 2];
          acc = __builtin_amdgcn_wmma_f32_16x16x4_f32(
              false, qA[kk], false, b, (short)0, acc, false, false);
        }
        // g = (1 - tanh(acc)^2) * W2[col]
        const float w2v = s.w2[hn + l16];
        float g[8];
        #pragma unroll
        for (int v = 0; v < 8; ++v) {
          float h = tanhf(acc[v]);
          g[v] = (1.0f - h * h) * w2v;
        }
        // In-wave 16x16 transpose C-layout -> A-layout via padded LDS scratch.
        __builtin_amdgcn_wave_barrier();
        #pragma unroll
        for (int v = 0; v < 8; ++v) scr[(hi16 * 8 + v) * 20 + l16] = g[v];
        __builtin_amdgcn_wave_barrier();
        v2f gf[4];
        #pragma unroll
        for (int f = 0; f < 4; ++f)
          gf[f] = *(const v2f*)&scr[l16 * 20 + f * 4 + hi16 * 2];
        __builtin_amdgcn_wave_barrier();
        // Backward: dVdq += g_tile @ W1^T[hn:hn+16, :]
        #pragma unroll
        for (int f = 0; f < 4; ++f) {
          #pragma unroll
          for (int dtn = 0; dtn < 8; ++dtn) {
            v2f b = *(const v2f*)&s.w1t[(nt * 4 + f) * 512 +
                                        (dtn * 16 + l16) * 4 + hi16 * 2];
            dacc[dtn] = __builtin_amdgcn_wmma_f32_16x16x4_f32(
                false, gf[f], false, b, (short)0, dacc[dtn], false, false);
          }
        }
      }
    }

    // Leapfrog: transpose dVdq tiles to A-layout; q += p_old*dt; p -= dVdq*dt.
    for (int dtn = 0; dtn < 8; ++dtn) {
      __builtin_amdgcn_wave_barrier();
      #pragma unroll
      for (int v = 0; v < 8; ++v) scr[(hi16 * 8 + v) * 20 + l16] = dacc[dtn][v];
      __builtin_amdgcn_wave_barrier();
      #pragma unroll
      for (int f = 0; f < 4; ++f) {
        v2f d2 = *(const v2f*)&scr[l16 * 20 + f * 4 + hi16 * 2];
        const int kk = dtn * 4 + f;
        float* pp = &pb[l16 * PSTRIDE + kk * 4 + hi16 * 2];
        v2f pv = *(const v2f*)pp;
        qA[kk] += DT * pv;          // uses old p
        pv -= DT * d2;
        *(v2f*)pp = pv;
      }
      __builtin_amdgcn_wave_barrier();
    }
  }

  float* outr = out + (size_t)row * (2 * ND);
  #pragma unroll
  for (int kk = 0; kk < 32; ++kk) {
    *(v2f*)(outr + kk * 4 + hi16 * 2) = qA[kk];
    v2f pv = *(const v2f*)&pb[l16 * PSTRIDE + kk * 4 + hi16 * 2];
    *(v2f*)(outr + ND + kk * 4 + hi16 * 2) = pv;
  }
}

extern "C" void kernel_launch(void* const* d_in, const int* in_sizes, int n_in,
                              void* d_out, int out_size, void* d_ws, size_t ws_size,
                              hipStream_t stream) {
  const float* x  = (const float*)d_in[0];
  const float* W1 = (const float*)d_in[1];
  const float* b1 = (const float*)d_in[2];
  const float* W2 = (const float*)d_in[3];
  // d_in[4] (b2) does not affect q/p dynamics.
  float* out = (float*)d_out;
  hipLaunchKernelGGL(hnode_kernel, dim3(B_TOTAL / ROWS_PER_BLOCK), dim3(256), 0,
                     stream, x, W1, b1, W2, out);
}